// GraphODEFunc_66580583023005
// MI455X (gfx1250) — compile-verified
//
#include <hip/hip_runtime.h>
#include <hip/hip_bf16.h>
#include <math.h>

// Problem constants (match reference: B=4, N=512, H=64)
#define GB 4
#define GN 512
#define GH 64

typedef __attribute__((ext_vector_type(2))) float v2f;
typedef __attribute__((ext_vector_type(8))) float v8f;

// ---------------------------------------------------------------------------
// Kernel 1: messages[b] = adjacency[b] (N x N) @ h[b] (N x H), fp32 WMMA.
// One wave (32 lanes) computes one 16x16 output tile.
//   A-frag (16x4 f32): lane L(0..15): rows M=L, K = k0 + {0,1}; lanes 16..31: K = k0 + {2,3}
//   B-frag (4x16 f32): v0: lanes 0-15 -> K=k0 row, lanes 16-31 -> K=k0+2 ; v1: K=k0+1 / k0+3
//   C/D  (16x16 f32): VGPR r: lanes 0-15 -> M=r, lanes 16-31 -> M=r+8 ; N=lane%16
// ---------------------------------------------------------------------------
__global__ __launch_bounds__(32) void msgs_gemm_kernel(
    const float* __restrict__ adj,   // (B, N, N)
    const float* __restrict__ h,     // (B, N, H)
    float* __restrict__ msg)         // (B, N, H)
{
    const int b  = blockIdx.z;
    const int m0 = blockIdx.x * 16;        // row tile within N
    const int n0 = blockIdx.y * 16;        // col tile within H
    const int lane = threadIdx.x;
    const int lr = lane & 15;              // 0..15
    const int hi = lane >> 4;              // 0 or 1 (upper K half / upper M half)

    const float* A  = adj + (size_t)b * GN * GN;
    const float* Bm = h   + (size_t)b * GN * GH;

    v8f acc = {};
    #pragma unroll 4
    for (int k0 = 0; k0 < GN; k0 += 4) {
        // A fragment: 2 contiguous fp32 per lane (8B aligned)
        const float* ap = A + (size_t)(m0 + lr) * GN + (k0 + 2 * hi);
        v2f afrag = *reinterpret_cast<const v2f*>(ap);
        // B fragment: rows k0+2*hi, k0+2*hi+1 of h, column n0+lr
        const float* bp = Bm + (size_t)(k0 + 2 * hi) * GH + (n0 + lr);
        v2f bfrag;
        bfrag.x = bp[0];
        bfrag.y = bp[GH];
        acc = __builtin_amdgcn_wmma_f32_16x16x4_f32(
            /*neg_a=*/false, afrag, /*neg_b=*/false, bfrag,
            /*c_mod=*/(short)0, acc, /*reuse_a=*/false, /*reuse_b=*/false);
    }

    float* out = msg + (size_t)b * GN * GH;
    #pragma unroll
    for (int r = 0; r < 8; ++r) {
        out[(size_t)(m0 + r + 8 * hi) * GH + (n0 + lr)] = acc[r];
    }
}

// ---------------------------------------------------------------------------
// Kernel 2: fused MLP on flattened rows (B*N = 2048 rows):
//   t1 = tanh([h | msg] @ W1 + b1)   (rows x 64), K = 128 split across h/msg
//   dh = t1 @ W2 + b2                (rows x 64), K = 64 via LDS-staged t1
// Block = 128 threads = 4 waves; block owns 16 rows; wave w owns N-tile 16*w.
// ---------------------------------------------------------------------------
__global__ __launch_bounds__(128) void mlp_fused_kernel(
    const float* __restrict__ h,     // (B*N, H)
    const float* __restrict__ msg,   // (B*N, H)
    const float* __restrict__ W1,    // (2H, H) row-major
    const float* __restrict__ b1,    // (H)
    const float* __restrict__ W2,    // (H, H) row-major
    const float* __restrict__ b2,    // (H)
    float* __restrict__ dh)          // (B*N, H)
{
    __shared__ float t1[16][68];     // 16 x 64 tile, padded stride 68 -> conflict-free

    const int r0   = blockIdx.x * 16;            // global row tile
    const int wave = threadIdx.x >> 5;           // 0..3
    const int lane = threadIdx.x & 31;
    const int lr   = lane & 15;
    const int hi   = lane >> 4;
    const int n0   = wave * 16;                  // output col tile

    // ---- GEMM1: acc1 = node_in(16 x 128) @ W1(128 x 16-tile) ----
    v8f acc1 = {};
    const float* srcs[2] = { h + (size_t)(r0 + lr) * GH,
                             msg + (size_t)(r0 + lr) * GH };
    #pragma unroll
    for (int half = 0; half < 2; ++half) {
        const float* arow = srcs[half];
        const float* Wbase = W1 + (size_t)(half * GH) * GH;
        #pragma unroll 4
        for (int k0 = 0; k0 < GH; k0 += 4) {
            v2f afrag = *reinterpret_cast<const v2f*>(arow + k0 + 2 * hi);
            const float* bp = Wbase + (size_t)(k0 + 2 * hi) * GH + (n0 + lr);
            v2f bfrag;
            bfrag.x = bp[0];
            bfrag.y = bp[GH];
            acc1 = __builtin_amdgcn_wmma_f32_16x16x4_f32(
                false, afrag, false, bfrag, (short)0, acc1, false, false);
        }
    }

    // ---- activation: tanh(acc + b1), stage 16x64 t1 tile into LDS ----
    const float bias1 = b1[n0 + lr];
    #pragma unroll
    for (int r = 0; r < 8; ++r) {
        t1[r + 8 * hi][n0 + lr] = tanhf(acc1[r] + bias1);
    }
    __syncthreads();

    // ---- GEMM2: acc2 = t1(16 x 64) @ W2(64 x 16-tile) ----
    v8f acc2 = {};
    #pragma unroll 4
    for (int k0 = 0; k0 < GH; k0 += 4) {
        v2f afrag = *reinterpret_cast<const v2f*>(&t1[lr][k0 + 2 * hi]);
        const float* bp = W2 + (size_t)(k0 + 2 * hi) * GH + (n0 + lr);
        v2f bfrag;
        bfrag.x = bp[0];
        bfrag.y = bp[GH];
        acc2 = __builtin_amdgcn_wmma_f32_16x16x4_f32(
            false, afrag, false, bfrag, (short)0, acc2, false, false);
    }

    const float bias2 = b2[n0 + lr];
    #pragma unroll
    for (int r = 0; r < 8; ++r) {
        dh[(size_t)(r0 + r + 8 * hi) * GH + (n0 + lr)] = acc2[r] + bias2;
    }
}

// ---------------------------------------------------------------------------
// Launch: input order = t, h, adjacency, W1, b1, W2, b2, E1, be1, E2, be2.
// Edge path (E1/be1/E2/be2) is dead in the reference -> skipped entirely.
// Workspace holds messages (B*N*H fp32 = 512 KB).
// ---------------------------------------------------------------------------
extern "C" void kernel_launch(void* const* d_in, const int* in_sizes, int n_in,
                              void* d_out, int out_size, void* d_ws, size_t ws_size,
                              hipStream_t stream) {
    const float* h   = (const float*)d_in[1];
    const float* adj = (const float*)d_in[2];
    const float* W1  = (const float*)d_in[3];
    const float* b1  = (const float*)d_in[4];
    const float* W2  = (const float*)d_in[5];
    const float* b2  = (const float*)d_in[6];
    float* msg = (float*)d_ws;                 // (B, N, H) scratch
    float* out = (float*)d_out;                // (B, N, H)

    dim3 g1(GN / 16, GH / 16, GB);             // 32 x 4 x 4 tiles
    msgs_gemm_kernel<<<g1, 32, 0, stream>>>(adj, h, msg);

    dim3 g2((GB * GN) / 16);                   // 128 row-tiles
    mlp_fused_kernel<<<g2, 128, 0, stream>>>(h, msg, W1, b1, W2, b2, out);
}